// SAGEConv_8177617732122
// MI455X (gfx1250) — compile-verified
//
#include <hip/hip_runtime.h>
#include <hip/hip_bf16.h>

typedef __attribute__((ext_vector_type(2))) float v2f;
typedef __attribute__((ext_vector_type(8))) float v8f;

#define BN_EPS 1e-5f
#define L2_EPS 1e-6f

// ---------------------------------------------------------------------------
// Kernel 1: gather self features + mean-aggregate neighbor features
//   Hcat[n, 0:D]   = feats[self_idx[n], :]
//   Hcat[n, D:2D]  = mean_s feats[neigh_idx[n,s], :]
// One block per output row; neighbor indices staged in LDS; float2 vectors.
// ---------------------------------------------------------------------------
__global__ __launch_bounds__(256) void sage_agg_kernel(
    const float* __restrict__ feats,
    const long long* __restrict__ self_idx,
    const long long* __restrict__ neigh_idx,
    float* __restrict__ Hcat,
    int N, int D, int S)
{
    __shared__ long long nidx[64];
    const int n = blockIdx.x;
    const int t = threadIdx.x;
    if (t < S) nidx[t] = neigh_idx[(long long)n * S + t];
    if (t == 0) nidx[63] = self_idx[n];
    __syncthreads();

    const float inv_s = 1.0f / (float)S;
    const long long si = nidx[63];
    const int D2 = D >> 1;                       // float2 columns
    const float2* __restrict__ f2 = (const float2*)feats;
    float2* __restrict__ orow = (float2*)(Hcat + (long long)n * (2 * D));

    for (int d = t; d < D2; d += blockDim.x) {
        orow[d] = f2[si * D2 + d];
        float ax = 0.0f, ay = 0.0f;
        #pragma unroll 5
        for (int s = 0; s < S; ++s) {
            float2 v = f2[nidx[s] * D2 + d];
            ax += v.x; ay += v.y;
        }
        orow[D2 + d] = make_float2(ax * inv_s, ay * inv_s);
    }
}

// ---------------------------------------------------------------------------
// Kernel 2: H = relu(Hcat @ W^T + b) via V_WMMA_F32_16X16X4_F32
//   A = Hcat [N x K] (K = 1024), W [NC x K] (NC = 512), H [N x NC]
// Block tile: BM=128 x BN=64, K tile BK=32. 256 threads = 8 waves.
// Each wave computes a 32x32 C patch = 2x2 tiles of 16x16 (4 x v8f accum).
//
// wave32 fragment layouts (CDNA5 ISA 7.12.2):
//   A 16x4 f32 : lanes 0-15 M=0..15 {VGPR0=K0, VGPR1=K1}; lanes 16-31 {K2,K3}
//   B 4x16 f32 : VGPR0 lanes0-15 K=0 (N=lane), lanes16-31 K=2; VGPR1 K=1/K=3
//   C 16x16    : VGPR r -> M=r (lanes 0-15), M=r+8 (lanes 16-31), N = lane%16
//
// B tile stored K-pair-packed in LDS (float2 per [k/2][n]) so each B
// fragment is a single aligned ds_load_b64 -> no v_mov reassembly.
// ---------------------------------------------------------------------------
#define BM 128
#define BN 64
#define BK 32

__global__ __launch_bounds__(256) void sage_gemm_relu_kernel(
    const float* __restrict__ A,     // Hcat  [N x K]
    const float* __restrict__ W,     // W     [NC x K]
    const float* __restrict__ bias,  // [NC]
    float* __restrict__ H,           // [N x NC]
    int N, int K, int NC)
{
    __shared__ float  As[BM][BK];        // row-major [m][k]
    __shared__ float2 Bs2[BK / 2][BN];   // [k-pair][n] = {B[2p][n], B[2p+1][n]}

    const int t    = threadIdx.x;
    const int lane = t & 31;
    const int wave = t >> 5;         // 0..7
    const int wm   = wave & 3;       // M pair index 0..3 (covers 8 M tiles)
    const int wn   = wave >> 2;      // N pair index 0..1 (covers 4 N tiles)
    const int half = lane >> 4;      // 0/1
    const int l16  = lane & 15;

    const int m_base = blockIdx.x * BM;
    const int n_base = blockIdx.y * BN;

    v8f acc[2][2] = {};

    for (int kt = 0; kt < K; kt += BK) {
        // --- load A tile: 128 rows x 32 cols, float4 per thread x4 ---
        #pragma unroll
        for (int it = 0; it < 4; ++it) {
            int lin  = t + 256 * it;        // 0..1023
            int row  = lin >> 3;            // 8 float4 per row
            int c4   = lin & 7;
            int grow = m_base + row;
            float4 v = make_float4(0.f, 0.f, 0.f, 0.f);
            if (grow < N)
                v = *(const float4*)(A + (long long)grow * K + kt + c4 * 4);
            *(float4*)(&As[row][c4 * 4]) = v;
        }
        // --- load B tile: W[n_base+j, kt+k] -> Bs2[k/2][j] (K-pair packed) ---
        #pragma unroll
        for (int it = 0; it < 2; ++it) {
            int lin  = t + 256 * it;        // 0..511
            int jrow = lin >> 3;            // 0..63
            int c4   = lin & 7;
            float4 v = *(const float4*)(W + (long long)(n_base + jrow) * K + kt + c4 * 4);
            Bs2[c4 * 2 + 0][jrow] = make_float2(v.x, v.y);
            Bs2[c4 * 2 + 1][jrow] = make_float2(v.z, v.w);
        }
        // prefetch next K tile of A (global_prefetch_b8)
        if (kt + BK < K) {
            int row  = t >> 3;
            int c4   = t & 7;
            int grow = m_base + row;
            if (grow < N)
                __builtin_prefetch(A + (long long)grow * K + kt + BK + c4 * 4, 0, 1);
        }
        __syncthreads();

        // --- 8 WMMA K-steps of 4 over the 32-wide K tile ---
        #pragma unroll
        for (int ks = 0; ks < BK; ks += 4) {
            v2f afr[2], bfr[2];
            #pragma unroll
            for (int i = 0; i < 2; ++i) {
                int m = (wm * 2 + i) * 16 + l16;
                afr[i].x = As[m][ks + 2 * half + 0];
                afr[i].y = As[m][ks + 2 * half + 1];
            }
            #pragma unroll
            for (int j = 0; j < 2; ++j) {
                int ncol = (wn * 2 + j) * 16 + l16;
                bfr[j] = *(const v2f*)&Bs2[(ks >> 1) + half][ncol];
            }
            #pragma unroll
            for (int i = 0; i < 2; ++i)
                #pragma unroll
                for (int j = 0; j < 2; ++j)
                    acc[i][j] = __builtin_amdgcn_wmma_f32_16x16x4_f32(
                        false, afr[i], false, bfr[j],
                        (short)0, acc[i][j], false, false);
        }
        __syncthreads();
    }

    // --- store C tiles with bias + ReLU ---
    #pragma unroll
    for (int i = 0; i < 2; ++i) {
        #pragma unroll
        for (int j = 0; j < 2; ++j) {
            int ncol = n_base + (wn * 2 + j) * 16 + l16;
            float bv = bias[ncol];
            #pragma unroll
            for (int r = 0; r < 8; ++r) {
                int m = m_base + (wm * 2 + i) * 16 + r + 8 * half;
                if (m < N) {
                    float v = acc[i][j][r] + bv;
                    H[(long long)m * NC + ncol] = v > 0.f ? v : 0.f;
                }
            }
        }
    }
}

// ---------------------------------------------------------------------------
// Kernel 3: per-column sum and sum-of-squares (for batch-norm statistics)
// stats[0:NC] = sum, stats[NC:2NC] = sumsq. Requires NC == 2*blockDim.x.
// ---------------------------------------------------------------------------
__global__ __launch_bounds__(256) void sage_stats_kernel(
    const float* __restrict__ H, float* __restrict__ stats,
    int N, int NC, int rows_per_block)
{
    const int t  = threadIdx.x;
    const int r0 = blockIdx.x * rows_per_block;
    int r1 = r0 + rows_per_block;
    if (r1 > N) r1 = N;

    float s0 = 0.f, s1 = 0.f, q0 = 0.f, q1 = 0.f;
    for (int r = r0; r < r1; ++r) {
        const float* row = H + (long long)r * NC;
        float a = row[t];
        float b = row[t + 256];
        s0 += a; q0 += a * a;
        s1 += b; q1 += b * b;
    }
    atomicAdd(&stats[t],            s0);
    atomicAdd(&stats[t + 256],      s1);
    atomicAdd(&stats[NC + t],       q0);
    atomicAdd(&stats[NC + t + 256], q1);
}

// ---------------------------------------------------------------------------
// Kernel 4: batch-norm (biased variance) + row-wise L2 normalize
// One block per row; each thread owns columns t and t+256.
// ---------------------------------------------------------------------------
__global__ __launch_bounds__(256) void sage_finalize_kernel(
    const float* __restrict__ H,
    const float* __restrict__ stats,
    const float* __restrict__ gamma,
    const float* __restrict__ beta,
    float* __restrict__ out,
    int N, int NC)
{
    __shared__ float red[8];
    const int n = blockIdx.x;
    const int t = threadIdx.x;
    const float invN = 1.0f / (float)N;
    const float* __restrict__ row = H + (long long)n * NC;

    float v[2];
    #pragma unroll
    for (int p = 0; p < 2; ++p) {
        int d = t + p * 256;
        float mu  = stats[d] * invN;
        float var = stats[NC + d] * invN - mu * mu;
        float x   = row[d];
        v[p] = (x - mu) * rsqrtf(var + BN_EPS) * gamma[d] + beta[d];
    }

    float ss = v[0] * v[0] + v[1] * v[1];
    #pragma unroll
    for (int o = 16; o > 0; o >>= 1)
        ss += __shfl_down(ss, o, 32);
    if ((t & 31) == 0) red[t >> 5] = ss;
    __syncthreads();
    float tot = 0.f;
    #pragma unroll
    for (int i = 0; i < 8; ++i) tot += red[i];

    float scale = 1.0f / (sqrtf(tot) + L2_EPS);
    float* __restrict__ orow = out + (long long)n * NC;
    orow[t]       = v[0] * scale;
    orow[t + 256] = v[1] * scale;
}

// ---------------------------------------------------------------------------
extern "C" void kernel_launch(void* const* d_in, const int* in_sizes, int n_in,
                              void* d_out, int out_size, void* d_ws, size_t ws_size,
                              hipStream_t stream) {
    const float*     feats     = (const float*)d_in[0];     // [N_TOTAL, D]
    const float*     W         = (const float*)d_in[1];     // [D, 2D]
    const float*     b         = (const float*)d_in[2];     // [D]
    const float*     gamma     = (const float*)d_in[3];     // [D]
    const float*     beta      = (const float*)d_in[4];     // [D]
    const long long* self_idx  = (const long long*)d_in[5]; // [N] int64
    const long long* neigh_idx = (const long long*)d_in[6]; // [N, S] int64

    const int N = in_sizes[5];          // 20000
    const int D = in_sizes[2];          // 512
    const int S = in_sizes[6] / N;      // 25
    const int K = 2 * D;                // 1024

    // workspace layout (floats): Hcat [N x 2D] | H [N x D] | stats [2D]
    float* Hcat  = (float*)d_ws;
    float* Hmid  = Hcat + (size_t)N * K;
    float* stats = Hmid + (size_t)N * D;

    hipMemsetAsync(stats, 0, (size_t)(2 * D) * sizeof(float), stream);

    sage_agg_kernel<<<N, 256, 0, stream>>>(feats, self_idx, neigh_idx, Hcat, N, D, S);

    dim3 ggrid((N + BM - 1) / BM, D / BN);
    sage_gemm_relu_kernel<<<ggrid, 256, 0, stream>>>(Hcat, W, b, Hmid, N, K, D);

    const int RPB = 100;
    sage_stats_kernel<<<(N + RPB - 1) / RPB, 256, 0, stream>>>(Hmid, stats, N, D, RPB);

    sage_finalize_kernel<<<N, 256, 0, stream>>>(Hmid, stats, gamma, beta,
                                                (float*)d_out, N, D);
}